// HierarchicalGraphConvolution_29283087024202
// MI455X (gfx1250) — compile-verified
//
#include <hip/hip_runtime.h>
#include <math.h>

typedef __attribute__((ext_vector_type(2))) float v2f;
typedef __attribute__((ext_vector_type(8))) float v8f;

#define NN 40000
#define EE 640000
#define FF 128

// ---------------------------------------------------------------------------
// Kernel 1: sem_att = softmax(sem_w); wscaled[k][n] = sem_att[k] * weight[k][n]
// Single block, 256 threads.
// ---------------------------------------------------------------------------
__global__ void k_sem_wscale(const float* __restrict__ sem_w,
                             const float* __restrict__ weight,
                             float* __restrict__ wscaled) {
    __shared__ float satt[FF];
    __shared__ float red[256];
    int t = threadIdx.x;
    float v = (t < FF) ? sem_w[t] : -INFINITY;
    red[t] = v;
    __syncthreads();
    for (int off = 128; off > 0; off >>= 1) {
        if (t < off) red[t] = fmaxf(red[t], red[t + off]);
        __syncthreads();
    }
    float mx = red[0];
    __syncthreads();
    float e = (t < FF) ? __expf(v - mx) : 0.0f;
    red[t] = e;
    __syncthreads();
    for (int off = 128; off > 0; off >>= 1) {
        if (t < off) red[t] += red[t + off];
        __syncthreads();
    }
    float inv = 1.0f / red[0];
    if (t < FF) satt[t] = e * inv;
    __syncthreads();
    for (int i = t; i < FF * FF; i += 256) {
        int k = i >> 7;
        wscaled[i] = satt[k] * weight[i];
    }
}

// ---------------------------------------------------------------------------
// Kernel 2: natt[i] = sigmoid(dot(x[i,:], node_w))  -- one wave32 per node
// ---------------------------------------------------------------------------
__global__ void k_node_att(const float* __restrict__ x,
                           const float* __restrict__ node_w,
                           float* __restrict__ natt) {
    int wave = (int)((blockIdx.x * blockDim.x + threadIdx.x) >> 5);
    int lane = threadIdx.x & 31;
    if (wave >= NN) return;
    const float4* xr = (const float4*)(x + (size_t)wave * FF);
    const float4* wr = (const float4*)node_w;
    float4 xv = xr[lane];
    float4 wv = wr[lane];
    float d = xv.x * wv.x + xv.y * wv.y + xv.z * wv.z + xv.w * wv.w;
    #pragma unroll
    for (int off = 16; off > 0; off >>= 1) d += __shfl_xor(d, off, 32);
    if (lane == 0) natt[wave] = 1.0f / (1.0f + __expf(-d));
}

// ---------------------------------------------------------------------------
// Kernel 3: support = natt ⊙ (x @ wscaled) via v_wmma_f32_16x16x4_f32.
// Block = 256 threads = 8 waves; block covers 16 rows x 128 cols.
// Wave w computes the 16x16 tile at cols [16w, 16w+16). K-loop: 32 WMMAs.
// A-operand (16x4 f32): lane L -> M = L%16; VGPR v -> K = 2*(L/16)+v.
// B-operand (4x16 f32): lane L -> N = L%16; VGPR v -> K = 2*(L/16)+v.
// C/D (16x16 f32):      lane L, VGPR r -> M = r + 8*(L/16), N = L%16.
// ---------------------------------------------------------------------------
__global__ void k_gemm_wmma(const float* __restrict__ x,
                            const float* __restrict__ wscaled,
                            const float* __restrict__ natt,
                            float* __restrict__ support) {
    __shared__ float lx[16 * FF];  // 8 KB A-strip

    const int row0 = blockIdx.x * 16;
    const int t    = threadIdx.x;
    const int wv   = t >> 5;        // wave id 0..7 -> col tile
    const int lane = t & 31;
    const int col0 = wv * 16;

    // Stage 16x128 A-strip into LDS: 512 float4, 2 per thread.
    {
        const float4* xsrc = (const float4*)(x + (size_t)row0 * FF);
        float4* ldst = (float4*)lx;
        ldst[t]       = xsrc[t];
        ldst[t + 256] = xsrc[t + 256];
    }
    __syncthreads();

    const int m = lane & 15;
    const int g = lane >> 4;
    const int nn = lane & 15;

    v8f acc = {};
    #pragma unroll 4
    for (int kt = 0; kt < 32; ++kt) {
        const int k0 = kt * 4;
        // A from LDS: two consecutive floats, 8-byte aligned -> ds_load_b64
        v2f a = *(const v2f*)&lx[m * FF + k0 + 2 * g];
        // B from global (L2-resident 64KB matrix)
        v2f b;
        const int kb = k0 + 2 * g;
        b.x = wscaled[(size_t)kb * FF + col0 + nn];
        b.y = wscaled[(size_t)(kb + 1) * FF + col0 + nn];
        acc = __builtin_amdgcn_wmma_f32_16x16x4_f32(
            /*neg_a=*/false, a, /*neg_b=*/false, b,
            /*c_mod=*/(short)0, acc, /*reuse_a=*/false, /*reuse_b=*/false);
    }

    // Row-scale by node attention and store.
    #pragma unroll
    for (int r = 0; r < 8; ++r) {
        const int row = row0 + r + 8 * g;
        support[(size_t)row * FF + col0 + nn] = acc[r] * natt[row];
    }
}

// ---------------------------------------------------------------------------
// Kernel 4: out[n, :] = bias[:]   (initialize accumulation target)
// ---------------------------------------------------------------------------
__global__ void k_init_out(const float* __restrict__ bias,
                           float* __restrict__ out) {
    int i = (int)(blockIdx.x * blockDim.x + threadIdx.x);  // float4 index
    const float4* b4 = (const float4*)bias;
    ((float4*)out)[i] = b4[i & 31];  // 32 float4 per row of 128
}

// ---------------------------------------------------------------------------
// Kernel 5: COO scatter-add. One wave32 per edge; float4 per lane;
// native global_atomic_add_f32 into L2-resident out.
// ---------------------------------------------------------------------------
__global__ void k_edges(const float* __restrict__ support,
                        const float* __restrict__ adj_val,
                        const int* __restrict__ src,
                        const int* __restrict__ dst,
                        float* __restrict__ out) {
    int e    = (int)((blockIdx.x * blockDim.x + threadIdx.x) >> 5);
    int lane = threadIdx.x & 31;
    if (e >= EE) return;
    int s = src[e];
    int d = dst[e];
    float a = adj_val[e];
    float4 sv = ((const float4*)(support + (size_t)s * FF))[lane];
    float* op = out + (size_t)d * FF + lane * 4;
    unsafeAtomicAdd(op + 0, sv.x * a);
    unsafeAtomicAdd(op + 1, sv.y * a);
    unsafeAtomicAdd(op + 2, sv.z * a);
    unsafeAtomicAdd(op + 3, sv.w * a);
}

// ---------------------------------------------------------------------------
extern "C" void kernel_launch(void* const* d_in, const int* in_sizes, int n_in,
                              void* d_out, int out_size, void* d_ws, size_t ws_size,
                              hipStream_t stream) {
    const float* x       = (const float*)d_in[0];
    const float* weight  = (const float*)d_in[1];
    const float* node_w  = (const float*)d_in[2];
    const float* sem_w   = (const float*)d_in[3];
    const float* bias    = (const float*)d_in[4];
    const float* adj_val = (const float*)d_in[5];
    const int*   src     = (const int*)d_in[6];
    const int*   dst     = (const int*)d_in[7];
    float* out = (float*)d_out;

    // Workspace layout (floats): wscaled[128*128] | natt[N] | support[N*128]
    float* ws       = (float*)d_ws;
    float* wscaled  = ws;                       // 16384
    float* natt     = wscaled + FF * FF;        // 40000
    float* support  = natt + NN;                // 5,120,000  (~20.7 MB total)

    // 1. softmax(sem_w) folded into weight
    k_sem_wscale<<<1, 256, 0, stream>>>(sem_w, weight, wscaled);
    // 2. per-node sigmoid attention (one wave per node)
    k_node_att<<<NN / 8, 256, 0, stream>>>(x, node_w, natt);
    // 3. WMMA GEMM: support = natt ⊙ (x @ wscaled)
    k_gemm_wmma<<<NN / 16, 256, 0, stream>>>(x, wscaled, natt, support);
    // 4. out = bias broadcast
    k_init_out<<<(NN * FF / 4) / 256, 256, 0, stream>>>(bias, out);
    // 5. edge scatter-add (one wave per edge)
    k_edges<<<EE / 8, 256, 0, stream>>>(support, adj_val, src, dst, out);
}